// AutoCorrelationLayer_82257213653395
// MI455X (gfx1250) — compile-verified
//
#include <hip/hip_runtime.h>
#include <hip/hip_bf16.h>

#define B_     32
#define L_     2048
#define E_     512
#define MODES_ 32
#define PI_F   3.14159265358979323846f

typedef __attribute__((ext_vector_type(2))) float v2f;
typedef __attribute__((ext_vector_type(8))) float v8f;

// D = A(16x4) * B(4x16) + C, fp32 WMMA
#define WMMA_F32(a, b, c) \
  __builtin_amdgcn_wmma_f32_16x16x4_f32(false, (a), false, (b), (short)0, (c), false, false)

// ---- WMMA fragment loaders (ISA 7.12.2 layouts, wave32) ----
// A 16x4 fp32: lanes 0-15 hold M=0..15 / K={k0,k0+1}; lanes 16-31 K={k0+2,k0+3}
__device__ __forceinline__ v2f lda_tile(const float* __restrict__ A, int ldA,
                                        int m0, int k0, int lane) {
  int m = m0 + (lane & 15);
  int k = k0 + ((lane >> 4) << 1);
  v2f r;
  r.x = A[(size_t)m * ldA + k];
  r.y = A[(size_t)m * ldA + k + 1];
  return r;
}
// B 4x16 fp32 row-major [K][N]
__device__ __forceinline__ v2f ldb_tile(const float* __restrict__ Bm, int ldB,
                                        int k0, int n0, int lane) {
  int n = n0 + (lane & 15);
  int k = k0 + ((lane >> 4) << 1);
  v2f r;
  r.x = Bm[(size_t)k * ldB + n];
  r.y = Bm[(size_t)(k + 1) * ldB + n];
  return r;
}
// B 4x16 fp32 loaded transposed from row-major W[N][K] (i.e. B = W^T)
__device__ __forceinline__ v2f ldb_tile_tr(const float* __restrict__ W, int ldW,
                                           int k0, int n0, int lane) {
  int n = n0 + (lane & 15);
  int k = k0 + ((lane >> 4) << 1);
  v2f r;
  r.x = W[(size_t)n * ldW + k];
  r.y = W[(size_t)n * ldW + k + 1];
  return r;
}
// C/D 16x16 fp32: VGPR r -> M = r + 8*(lane>=16), N = lane&15
__device__ __forceinline__ void stc_tile(float* __restrict__ C, int ldC,
                                         int m0, int n0, int lane, v8f acc) {
  int n  = n0 + (lane & 15);
  int mb = m0 + ((lane >> 4) << 3);
#pragma unroll
  for (int r = 0; r < 8; ++r) C[(size_t)(mb + r) * ldC + n] = acc[r];
}

// ---- K0: trig tables. Fc/FsN: [MODES][L] fwd basis (cos, -sin);
//          Finv: [L][2*MODES]  (cos | sin) inverse basis ----
__global__ void k_tables(float* __restrict__ Fc, float* __restrict__ FsN,
                         float* __restrict__ Finv) {
  int idx = blockIdx.x * blockDim.x + threadIdx.x;
  if (idx < L_ * 2 * MODES_) {
    int l = idx / (2 * MODES_);
    int k = idx % (2 * MODES_);
    int m = k & (MODES_ - 1);
    int r = (l * m) & (L_ - 1);
    float ang = (2.0f * PI_F / (float)L_) * (float)r;
    Finv[idx] = (k < MODES_) ? cosf(ang) : sinf(ang);
  }
  if (idx < MODES_ * L_) {
    int m = idx / L_;
    int l = idx % L_;
    int r = (l * m) & (L_ - 1);
    float ang = (2.0f * PI_F / (float)L_) * (float)r;
    Fc[idx]  = cosf(ang);
    FsN[idx] = -sinf(ang);
  }
}

// ---- K1: repack W1 (E,E,MODES) -> (MODES,E,E) for coalesced GEMM B loads ----
__global__ void k_transpose_w1(const float* __restrict__ W1r, const float* __restrict__ W1i,
                               float* __restrict__ W1rt, float* __restrict__ W1it) {
  int io = blockIdx.x * blockDim.x + threadIdx.x;  // flat (i,o)
  if (io >= E_ * E_) return;
  size_t src = (size_t)io * MODES_;
#pragma unroll
  for (int m = 0; m < MODES_; ++m) {
    W1rt[(size_t)m * E_ * E_ + io] = W1r[src + m];
    W1it[(size_t)m * E_ * E_ + io] = W1i[src + m];
  }
}

// ---- K2: forward 32-mode DFT of raw queries (single pass over queries).
//  PR[b,m,e] = sum_l cos(2pi m l/L) * queries[b,l,e]
//  PI[b,m,e] = sum_l -sin(2pi m l/L) * queries[b,l,e]
//  Each wave covers BOTH 16-row mode tiles so queries is read exactly once.
__global__ void k_fwd_dft(const float* __restrict__ queries, const float* __restrict__ Fc,
                          const float* __restrict__ FsN,
                          float* __restrict__ PR, float* __restrict__ PI) {
  const int NT = E_ / 16;       // 32
  int lane = threadIdx.x & 31;
  int wid  = blockIdx.x * (blockDim.x >> 5) + (threadIdx.x >> 5);
  int b    = wid / NT;
  if (b >= B_) return;
  int nt = wid % NT;
  const float* Q = queries + (size_t)b * L_ * E_;
  v8f accR0 = {}, accI0 = {}, accR1 = {}, accI1 = {};
#pragma unroll 4
  for (int k = 0; k < L_; k += 4) {
    v2f bQ  = ldb_tile(Q,   E_, k, nt * 16, lane);   // shared across all 4 WMMAs
    v2f aC0 = lda_tile(Fc,  L_, 0,  k, lane);
    v2f aS0 = lda_tile(FsN, L_, 0,  k, lane);
    v2f aC1 = lda_tile(Fc,  L_, 16, k, lane);
    v2f aS1 = lda_tile(FsN, L_, 16, k, lane);
    accR0 = WMMA_F32(aC0, bQ, accR0);
    accI0 = WMMA_F32(aS0, bQ, accI0);
    accR1 = WMMA_F32(aC1, bQ, accR1);
    accI1 = WMMA_F32(aS1, bQ, accI1);
  }
  float* pr = PR + (size_t)b * MODES_ * E_;
  float* pi = PI + (size_t)b * MODES_ * E_;
  stc_tile(pr, E_, 0,  nt * 16, lane, accR0);
  stc_tile(pi, E_, 0,  nt * 16, lane, accI0);
  stc_tile(pr, E_, 16, nt * 16, lane, accR1);
  stc_tile(pi, E_, 16, nt * 16, lane, accI1);
}

// ---- K3: sel = P @ Wq^T (+ bias folded into DC mode of real part) ----
__global__ void k_proj_q(const float* __restrict__ PR, const float* __restrict__ PI,
                         const float* __restrict__ Wq, const float* __restrict__ bq,
                         float* __restrict__ selR, float* __restrict__ selI) {
  const int NT = E_ / 16;                // 32
  const int MT = (B_ * MODES_) / 16;     // 64
  int lane = threadIdx.x & 31;
  int wid  = blockIdx.x * (blockDim.x >> 5) + (threadIdx.x >> 5);
  int mt = wid / NT;
  if (mt >= MT) return;
  int nt = wid % NT;
  v8f accR = {}; v8f accI = {};
#pragma unroll 4
  for (int k = 0; k < E_; k += 4) {
    v2f ar = lda_tile(PR, E_, mt * 16, k, lane);
    v2f ai = lda_tile(PI, E_, mt * 16, k, lane);
    v2f bw = ldb_tile_tr(Wq, E_, k, nt * 16, lane);
    accR = WMMA_F32(ar, bw, accR);
    accI = WMMA_F32(ai, bw, accI);
  }
  int n  = nt * 16 + (lane & 15);
  int mb = mt * 16 + ((lane >> 4) << 3);
  float biasDC = (float)L_ * bq[n];   // rfft of constant bias hits only mode 0
#pragma unroll
  for (int r = 0; r < 8; ++r) {
    int row = mb + r;                  // row = b*MODES + m
    float vR = accR[r];
    if ((row & (MODES_ - 1)) == 0) vR += biasDC;
    selR[(size_t)row * E_ + n] = vR;
    selI[(size_t)row * E_ + n] = accI[r];
  }
}

// ---- K4: per-mode complex mix  X_m = sel_m @ W1[:,:,m]  (complex GEMM).
//  Each wave covers BOTH 16-row batch tiles so W1 slices are read once.
__global__ void k_mode_mix(const float* __restrict__ selR, const float* __restrict__ selI,
                           const float* __restrict__ W1rt, const float* __restrict__ W1it,
                           float* __restrict__ XR, float* __restrict__ XI) {
  const int NT = E_ / 16;   // 32
  int lane = threadIdx.x & 31;
  int wid  = blockIdx.x * (blockDim.x >> 5) + (threadIdx.x >> 5);
  int m = wid / NT;
  if (m >= MODES_) return;
  int nt = wid % NT;
  const float* Ar = selR + (size_t)m * E_;           // rows = batch, stride MODES*E
  const float* Ai = selI + (size_t)m * E_;
  const float* Br = W1rt + (size_t)m * E_ * E_;      // row-major [i][o]
  const float* Bi = W1it + (size_t)m * E_ * E_;
  v8f aRP0 = {}, aRN0 = {}, aIm0 = {};
  v8f aRP1 = {}, aRN1 = {}, aIm1 = {};
#pragma unroll 2
  for (int k = 0; k < E_; k += 4) {
    v2f br  = ldb_tile(Br, E_, k, nt * 16, lane);    // shared across batch tiles
    v2f bi  = ldb_tile(Bi, E_, k, nt * 16, lane);
    v2f ar0 = lda_tile(Ar, MODES_ * E_, 0,  k, lane);
    v2f ai0 = lda_tile(Ai, MODES_ * E_, 0,  k, lane);
    v2f ar1 = lda_tile(Ar, MODES_ * E_, 16, k, lane);
    v2f ai1 = lda_tile(Ai, MODES_ * E_, 16, k, lane);
    aRP0 = WMMA_F32(ar0, br, aRP0);   // sR*wR
    aRN0 = WMMA_F32(ai0, bi, aRN0);   // sI*wI (subtract at epilogue)
    aIm0 = WMMA_F32(ar0, bi, aIm0);   // sR*wI
    aIm0 = WMMA_F32(ai0, br, aIm0);   // + sI*wR
    aRP1 = WMMA_F32(ar1, br, aRP1);
    aRN1 = WMMA_F32(ai1, bi, aRN1);
    aIm1 = WMMA_F32(ar1, bi, aIm1);
    aIm1 = WMMA_F32(ai1, br, aIm1);
  }
  int n = nt * 16 + (lane & 15);
  int h = ((lane >> 4) << 3);
#pragma unroll
  for (int r = 0; r < 8; ++r) {
    int b0 = h + r;        // batch tile 0: b = 0..15
    int b1 = 16 + h + r;   // batch tile 1: b = 16..31
    XR[((size_t)b0 * MODES_ + m) * E_ + n] = aRP0[r] - aRN0[r];
    XI[((size_t)b0 * MODES_ + m) * E_ + n] = aIm0[r];
    XR[((size_t)b1 * MODES_ + m) * E_ + n] = aRP1[r] - aRN1[r];
    XI[((size_t)b1 * MODES_ + m) * E_ + n] = aIm1[r];
  }
}

// ---- K5: Y = scale(X @ Wo^T) packed as (B, 64, E): rows 0..31 = c_m*YR,
//          rows 32..63 = -c_m*YI, c_0 = 1/L, c_m = 2/L  ----
__global__ void k_proj_o(const float* __restrict__ XR, const float* __restrict__ XI,
                         const float* __restrict__ Wo, float* __restrict__ Y) {
  const int NT = E_ / 16;               // 32
  const int MT = (B_ * MODES_) / 16;    // 64
  int lane = threadIdx.x & 31;
  int wid  = blockIdx.x * (blockDim.x >> 5) + (threadIdx.x >> 5);
  int mt = wid / NT;
  if (mt >= MT) return;
  int nt = wid % NT;
  v8f accR = {}, accI = {};
#pragma unroll 4
  for (int k = 0; k < E_; k += 4) {
    v2f ar = lda_tile(XR, E_, mt * 16, k, lane);
    v2f ai = lda_tile(XI, E_, mt * 16, k, lane);
    v2f bw = ldb_tile_tr(Wo, E_, k, nt * 16, lane);
    accR = WMMA_F32(ar, bw, accR);
    accI = WMMA_F32(ai, bw, accI);
  }
  int n  = nt * 16 + (lane & 15);
  int mb = mt * 16 + ((lane >> 4) << 3);
#pragma unroll
  for (int r = 0; r < 8; ++r) {
    int row = mb + r;
    int b = row >> 5;
    int m = row & (MODES_ - 1);
    float c = ((m == 0) ? 1.0f : 2.0f) / (float)L_;
    Y[((size_t)b * 64 + m) * E_ + n]      =  c * accR[r];
    Y[((size_t)b * 64 + 32 + m) * E_ + n] = -c * accI[r];
  }
}

// ---- K6: inverse transform + bias: out[b,l,o] = Finv[l,:] . Y[b,:,o] + bo[o].
//  Each wave does a 16(l) x 64(o) strip: Finv A-tile amortized over 4 N tiles.
__global__ void k_inv(const float* __restrict__ Finv, const float* __restrict__ Y,
                      const float* __restrict__ bo, float* __restrict__ out) {
  const int LT  = L_ / 16;   // 128
  const int NT4 = E_ / 64;   // 8 strips of 4 n-tiles
  int lane = threadIdx.x & 31;
  int wid  = blockIdx.x * (blockDim.x >> 5) + (threadIdx.x >> 5);
  int b = wid / (LT * NT4);
  if (b >= B_) return;
  int rem = wid % (LT * NT4);
  int lt = rem / NT4;
  int n0 = (rem % NT4) * 64;
  const float* Yb = Y + (size_t)b * 64 * E_;
  v8f acc0 = {}, acc1 = {}, acc2 = {}, acc3 = {};
#pragma unroll
  for (int k = 0; k < 64; k += 4) {
    v2f a   = lda_tile(Finv, 2 * MODES_, lt * 16, k, lane);  // shared across 4 n-tiles
    v2f bb0 = ldb_tile(Yb, E_, k, n0,      lane);
    v2f bb1 = ldb_tile(Yb, E_, k, n0 + 16, lane);
    v2f bb2 = ldb_tile(Yb, E_, k, n0 + 32, lane);
    v2f bb3 = ldb_tile(Yb, E_, k, n0 + 48, lane);
    acc0 = WMMA_F32(a, bb0, acc0);
    acc1 = WMMA_F32(a, bb1, acc1);
    acc2 = WMMA_F32(a, bb2, acc2);
    acc3 = WMMA_F32(a, bb3, acc3);
  }
  int nl = lane & 15;
  int mb = lt * 16 + ((lane >> 4) << 3);
  float* O = out + (size_t)b * L_ * E_;
#pragma unroll
  for (int j = 0; j < 4; ++j) {
    int n = n0 + j * 16 + nl;
    float bias = bo[n];
    v8f acc = (j == 0) ? acc0 : (j == 1) ? acc1 : (j == 2) ? acc2 : acc3;
#pragma unroll
    for (int r = 0; r < 8; ++r) O[(size_t)(mb + r) * E_ + n] = acc[r] + bias;
  }
}

extern "C" void kernel_launch(void* const* d_in, const int* in_sizes, int n_in,
                              void* d_out, int out_size, void* d_ws, size_t ws_size,
                              hipStream_t stream) {
  const float* queries = (const float*)d_in[0];
  const float* Wq      = (const float*)d_in[1];
  const float* bq      = (const float*)d_in[2];
  const float* W1r     = (const float*)d_in[3];
  const float* W1i     = (const float*)d_in[4];
  const float* Wo      = (const float*)d_in[5];
  const float* bo      = (const float*)d_in[6];
  float* out = (float*)d_out;

  // workspace carve-up (floats)
  float* ws   = (float*)d_ws;
  float* Fc   = ws;                               // MODES*L =   65536
  float* FsN  = Fc  + (size_t)MODES_ * L_;
  float* Finv = FsN + (size_t)MODES_ * L_;        // L*64    =  131072
  float* W1rt = Finv + (size_t)L_ * 2 * MODES_;   // 8388608
  float* W1it = W1rt + (size_t)MODES_ * E_ * E_;  // 8388608
  float* PR   = W1it + (size_t)MODES_ * E_ * E_;  // 524288 each below
  float* PI_  = PR   + (size_t)B_ * MODES_ * E_;
  float* selR = PI_  + (size_t)B_ * MODES_ * E_;
  float* selI = selR + (size_t)B_ * MODES_ * E_;
  float* XR   = selI + (size_t)B_ * MODES_ * E_;
  float* XI   = XR   + (size_t)B_ * MODES_ * E_;
  float* Y    = XI   + (size_t)B_ * MODES_ * E_;  // B*64*E = 1048576

  const int TPB = 256;
  const int WPB = TPB / 32;

  // tables + W1 repack
  k_tables<<<(L_ * 2 * MODES_ + TPB - 1) / TPB, TPB, 0, stream>>>(Fc, FsN, Finv);
  k_transpose_w1<<<(E_ * E_ + TPB - 1) / TPB, TPB, 0, stream>>>(W1r, W1i, W1rt, W1it);

  // forward 32-mode DFT: B * (E/16) = 1024 waves (queries read exactly once)
  {
    int waves = B_ * (E_ / 16);
    k_fwd_dft<<<(waves + WPB - 1) / WPB, TPB, 0, stream>>>(queries, Fc, FsN, PR, PI_);
  }
  // sel = P @ Wq^T + bias@DC : 64*32 = 2048 waves
  {
    int waves = ((B_ * MODES_) / 16) * (E_ / 16);
    k_proj_q<<<(waves + WPB - 1) / WPB, TPB, 0, stream>>>(PR, PI_, Wq, bq, selR, selI);
  }
  // per-mode complex mix: 32 * 32 = 1024 waves (W1 read exactly once)
  {
    int waves = MODES_ * (E_ / 16);
    k_mode_mix<<<(waves + WPB - 1) / WPB, TPB, 0, stream>>>(selR, selI, W1rt, W1it, XR, XI);
  }
  // Wo projection + irfft scaling pack: 2048 waves
  {
    int waves = ((B_ * MODES_) / 16) * (E_ / 16);
    k_proj_o<<<(waves + WPB - 1) / WPB, TPB, 0, stream>>>(XR, XI, Wo, Y);
  }
  // inverse transform + bias: B * (L/16) * (E/64) = 32768 waves
  {
    int waves = B_ * (L_ / 16) * (E_ / 64);
    k_inv<<<(waves + WPB - 1) / WPB, TPB, 0, stream>>>(Finv, Y, bo, out);
  }
}